// GIN_2layer_79027398246922
// MI455X (gfx1250) — compile-verified
//
#include <hip/hip_runtime.h>

// ---------------------------------------------------------------------------
// GIN 2-layer GNN for MI455X (gfx1250, wave32).
//   z  = x + scatter_add(x[src] -> dst)           (f32 HW atomics, L2 resident)
//   h  = relu(z @ W1^T + b1)                      (V_WMMA_F32_16X16X4_F32)
//   z2 = h + scatter_add(h[src] -> dst)
//   out= z2 @ W2^T + b2                           (V_WMMA_F32_16X16X4_F32)
// ---------------------------------------------------------------------------

typedef float v2f __attribute__((ext_vector_type(2)));
typedef float v8f __attribute__((ext_vector_type(8)));

#define N_NODES   100000
#define N_EDGES   3200000
#define D_IN      128
#define D_HID     128
#define N_CLASSES 64

// ---- elementwise copy (z := x), float4 vectorized --------------------------
__global__ __launch_bounds__(256) void gin_copy_kernel(
    const float* __restrict__ src, float* __restrict__ dst, int n4) {
  int i = blockIdx.x * blockDim.x + threadIdx.x;
  if (i < n4) {
    ((float4*)dst)[i] = ((const float4*)src)[i];
  }
}

// ---- edge scatter-add: one wave32 per edge, one lane per 4 features --------
// Gather of feat[src] is a fully-coalesced 512B row per wave (all L2-resident);
// accumulation uses hardware global_atomic_add_f32 (no CAS loop), contiguous
// per-row so atomics pack per cacheline.
__global__ __launch_bounds__(256) void gin_scatter_add_kernel(
    const float* __restrict__ feat,
    const int*   __restrict__ srcIdx,
    const int*   __restrict__ dstIdx,
    float*       __restrict__ acc) {
  int gid = blockIdx.x * blockDim.x + threadIdx.x;
  int e = gid >> 5;          // edge id (wave-uniform)
  int q = (gid & 31) << 2;   // feature offset 0..124, step 4
  if (e >= N_EDGES) return;
  int s = srcIdx[e];
  int d = dstIdx[e];
  const float4 v = *(const float4*)(feat + (size_t)s * D_IN + q);
  float* p = acc + (size_t)d * D_IN + q;
  unsafeAtomicAdd(p + 0, v.x);   // -> global_atomic_add_f32
  unsafeAtomicAdd(p + 1, v.y);
  unsafeAtomicAdd(p + 2, v.z);
  unsafeAtomicAdd(p + 3, v.w);
}

// ---- node MLP via f32 WMMA -------------------------------------------------
// OUT[m][n] = (relu?)( sum_k Z[m][k] * W[n][k] + B[n] ),  K = D_IN = 128.
// One wave per 16-node M tile; 8 waves (256 thr) per block = 128 nodes/block.
// A tile 16x4 f32: lane = m + 16*(k>>1), vgpr = k&1  -> contiguous v2f load.
// B tile 4x16 f32: lane = n + 16*(k>>1), vgpr = k&1  -> contiguous v2f load.
// C/D 16x16 f32:   vgpr r holds M = r + 8*(lane>>4), N = lane&15.
template <int NOUT, bool RELU>
__global__ __launch_bounds__(256) void gin_mlp_wmma_kernel(
    const float* __restrict__ Z,    // [N_NODES x D_IN]
    const float* __restrict__ W,    // [NOUT x D_IN] row-major
    const float* __restrict__ Bv,   // [NOUT]
    float*       __restrict__ OUT)  // [N_NODES x NOUT]
{
  const int lane   = threadIdx.x & 31;
  const int wave   = threadIdx.x >> 5;
  const int m_base = (blockIdx.x * 8 + wave) * 16;
  if (m_base >= N_NODES) return;           // wave-uniform: EXEC stays all-ones

  const int half = lane >> 4;              // 0 | 1  (K sub-pair select)
  const int l16  = lane & 15;

  const float* arow = Z + (size_t)(m_base + l16) * D_IN + 2 * half;

#pragma unroll
  for (int nb = 0; nb < NOUT / 16; ++nb) {
    const int n_base = nb * 16;
    const float* brow = W + (size_t)(n_base + l16) * D_IN + 2 * half;

    const float bias = Bv[n_base + l16];   // C(m,n) = bias[n], same for all M
    v8f acc;
#pragma unroll
    for (int r = 0; r < 8; ++r) acc[r] = bias;

#pragma unroll 8
    for (int k0 = 0; k0 < D_IN; k0 += 4) {
      v2f a = *(const v2f*)(arow + k0);
      v2f b = *(const v2f*)(brow + k0);
      // D = A x B + C, f32 in / f32 accumulate (exact vs reference)
      acc = __builtin_amdgcn_wmma_f32_16x16x4_f32(
          /*neg_a=*/false, a, /*neg_b=*/false, b,
          /*c_mod=*/(short)0, acc, /*reuse_a=*/false, /*reuse_b=*/false);
    }

    float* orow = OUT + (size_t)(m_base + 8 * half) * NOUT + n_base + l16;
#pragma unroll
    for (int r = 0; r < 8; ++r) {
      float v = acc[r];
      if (RELU) v = fmaxf(v, 0.0f);
      orow[(size_t)r * NOUT] = v;
    }
  }
}

// ---------------------------------------------------------------------------
extern "C" void kernel_launch(void* const* d_in, const int* in_sizes, int n_in,
                              void* d_out, int out_size, void* d_ws, size_t ws_size,
                              hipStream_t stream) {
  const float* x  = (const float*)d_in[0];
  const int*   ei = (const int*)d_in[1];      // [2 x N_EDGES], row0=src row1=dst
  const float* W1 = (const float*)d_in[2];
  const float* b1 = (const float*)d_in[3];
  const float* W2 = (const float*)d_in[4];
  const float* b2 = (const float*)d_in[5];
  const int* src = ei;
  const int* dst = ei + N_EDGES;

  float* z = (float*)d_ws;                       // [N_NODES x 128]  51.2 MB
  float* h = z + (size_t)N_NODES * D_IN;         // [N_NODES x 128]  51.2 MB

  const int feat4       = N_NODES * D_IN / 4;               // float4 count
  const int copy_blocks = (feat4 + 255) / 256;
  const int scat_blocks = (int)(((long long)N_EDGES * 32 + 255) / 256); // 400000
  const int gemm_blocks = (N_NODES / 16 + 7) / 8;                       // 782

  // ---- layer 1 ----
  gin_copy_kernel<<<copy_blocks, 256, 0, stream>>>(x, z, feat4);         // z = x
  gin_scatter_add_kernel<<<scat_blocks, 256, 0, stream>>>(x, src, dst, z);
  gin_mlp_wmma_kernel<D_HID, true><<<gemm_blocks, 256, 0, stream>>>(z, W1, b1, h);

  // ---- layer 2 ----
  gin_copy_kernel<<<copy_blocks, 256, 0, stream>>>(h, z, feat4);         // z = h
  gin_scatter_add_kernel<<<scat_blocks, 256, 0, stream>>>(h, src, dst, z);
  gin_mlp_wmma_kernel<N_CLASSES, false><<<gemm_blocks, 256, 0, stream>>>(
      z, W2, b2, (float*)d_out);
}